// GCN_62818191671845
// MI455X (gfx1250) — compile-verified
//
#include <hip/hip_runtime.h>
#include <math.h>

// Problem constants (match reference setup_inputs exactly)
#define NGRAPH 32
#define NPG    2048
#define NNODE  (NGRAPH * NPG)     // 65536
#define DEG    16
#define NEDGE  (NNODE * DEG)      // 1048576
#define FIN0   16
#define HDIM   64
#define CDIM   8
#define BNEPS  1e-5f
#define NEGINF (-3.402823466e38f)

typedef float v2f __attribute__((ext_vector_type(2)));
typedef float v8f __attribute__((ext_vector_type(8)));

// ---------------------------------------------------------------------------
// init: mask = 1.0 for all nodes, edge-valid = 1 for all edges
// ---------------------------------------------------------------------------
__global__ void init_state(float* __restrict__ maskA, unsigned char* __restrict__ ev) {
    int t = blockIdx.x * blockDim.x + threadIdx.x;
    if (t < NNODE) maskA[t] = 1.0f;
    if (t < NEDGE) ev[t] = 1;
}

// ---------------------------------------------------------------------------
// Masked BatchNorm statistics: stats[0..F) = sum(x*m), stats[F..2F) = sum(x^2*m),
// stats[2F] = sum(m).  stats must be zeroed before launch.
// blockDim = 256, F in {16,64}.
// ---------------------------------------------------------------------------
__global__ __launch_bounds__(256) void bn_stats(const float* __restrict__ x,
                                                const float* __restrict__ maskf,
                                                int F, float* __restrict__ stats) {
    __shared__ float s1[256], s2[256], snn[32];
    const int f = threadIdx.x % F;
    const int g = threadIdx.x / F;
    const int G = 256 / F;
    float a = 0.f, b = 0.f, n = 0.f;
    for (int node = blockIdx.x * G + g; node < NNODE; node += gridDim.x * G) {
        float m = maskf[node];
        float v = x[(size_t)node * F + f];
        a += v * m;
        b += v * v * m;
        if (f == 0) n += m;
    }
    s1[threadIdx.x] = a;
    s2[threadIdx.x] = b;
    if (f == 0) snn[g] = n;
    __syncthreads();
    if (g == 0) {
        float ta = 0.f, tb = 0.f;
        for (int gg = 0; gg < G; ++gg) { ta += s1[f + gg * F]; tb += s2[f + gg * F]; }
        atomicAdd(&stats[f], ta);
        atomicAdd(&stats[F + f], tb);
    }
    if (threadIdx.x == 0) {
        float tn = 0.f;
        for (int gg = 0; gg < G; ++gg) tn += snn[gg];
        atomicAdd(&stats[2 * F], tn);
    }
}

// ---------------------------------------------------------------------------
// BN apply: xn = ((x - mu) * rsqrt(var+eps) * g + b) * mask
// fshift = log2(F)
// ---------------------------------------------------------------------------
__global__ void bn_apply(const float* __restrict__ x, const float* __restrict__ maskf,
                         const float* __restrict__ gam, const float* __restrict__ bet,
                         const float* __restrict__ stats, int fshift,
                         float* __restrict__ xn) {
    const int F = 1 << fshift;
    long long t = (long long)blockIdx.x * blockDim.x + threadIdx.x;
    if (t >= (long long)NNODE * F) return;
    int node = (int)(t >> fshift);
    int f = (int)(t & (F - 1));
    float n = stats[2 * F];
    float mu = stats[f] / n;
    float var = stats[F + f] / n - mu * mu;
    float rs = rsqrtf(var + BNEPS);
    float v = x[t];
    xn[t] = ((v - mu) * rs * gam[f] + bet[f]) * maskf[node];
}

// ---------------------------------------------------------------------------
// Edge scatter: agg[dst] += xn[src]  (valid edges only), cnt[dst] += 1.
// One thread per (edge, feature). fshift = log2(F).
// ---------------------------------------------------------------------------
__global__ void sage_scatter(const float* __restrict__ xn, const int* __restrict__ src,
                             const int* __restrict__ dst, const unsigned char* __restrict__ ev,
                             int fshift, float* __restrict__ agg, float* __restrict__ cntv) {
    const int F = 1 << fshift;
    long long t = (long long)blockIdx.x * blockDim.x + threadIdx.x;
    if (t >= (long long)NEDGE * F) return;
    int e = (int)(t >> fshift);
    int f = (int)(t & (F - 1));
    if (!ev[e]) return;
    int s = src[e], d = dst[e];
    unsafeAtomicAdd(&agg[((size_t)d << fshift) + f], xn[((size_t)s << fshift) + f]);
    if (f == 0) unsafeAtomicAdd(&cntv[d], 1.0f);
}

// ---------------------------------------------------------------------------
// WMMA GEMM:  out[i,:] = (agg[i,:]/max(cnt[i],1)) @ Wl + xn[i,:] @ Wr + bias,
// optional ReLU.  A: [16 x Fin] f32, B: [Fin x 64] f32 row-major, chained
// v_wmma_f32_16x16x4_f32.  grid.x = NNODE/16, blockDim = 128 (4 waves = 4
// column tiles of the 64-wide output).  No divergence -> EXEC all-ones.
//
// f32 fragment layouts per CDNA5 ISA 7.12.2:
//  A 16x4:  lanes 0-15 hold {K=k,k+1} of row M=lane; lanes 16-31 hold {k+2,k+3}
//  B 4x16:  VGPR v: lanes 0-15 = row K=k+v, lanes 16-31 = row K=k+2+v; N = lane%16
//  C/D:     VGPR v: lanes 0-15 -> M=v, lanes 16-31 -> M=v+8; N = lane%16
// ---------------------------------------------------------------------------
__global__ __launch_bounds__(128) void sage_gemm_wmma(
    const float* __restrict__ agg, const float* __restrict__ cntv,
    const float* __restrict__ xn, const float* __restrict__ Wl,
    const float* __restrict__ Wr, const float* __restrict__ bias,
    int Fin, int do_relu, float* __restrict__ out) {
    const int lane = threadIdx.x & 31;
    const int wv   = threadIdx.x >> 5;     // column tile 0..3
    const int half = lane >> 4;            // 0: K lo pair, 1: K hi pair
    const int l    = lane & 15;
    const int row  = blockIdx.x * 16 + l;  // A-fragment row for this lane
    const int n0   = wv * 16;

    const float invc = 1.0f / fmaxf(cntv[row], 1.0f);
    const float* arowL = agg + (size_t)row * Fin;
    const float* arowR = xn  + (size_t)row * Fin;

    v8f acc = {0.f, 0.f, 0.f, 0.f, 0.f, 0.f, 0.f, 0.f};
    for (int k = 0; k < Fin; k += 4) {
        const int ka = k + 2 * half;
        v2f aL, aR, bL, bR;
        aL.x = arowL[ka] * invc;
        aL.y = arowL[ka + 1] * invc;
        aR.x = arowR[ka];
        aR.y = arowR[ka + 1];
        bL.x = Wl[(size_t)ka * HDIM + n0 + l];
        bL.y = Wl[(size_t)(ka + 1) * HDIM + n0 + l];
        bR.x = Wr[(size_t)ka * HDIM + n0 + l];
        bR.y = Wr[(size_t)(ka + 1) * HDIM + n0 + l];
        acc = __builtin_amdgcn_wmma_f32_16x16x4_f32(false, aL, false, bL,
                                                    (short)0, acc, false, false);
        acc = __builtin_amdgcn_wmma_f32_16x16x4_f32(false, aR, false, bR,
                                                    (short)0, acc, false, false);
    }
    const float bb = bias[n0 + l];
#pragma unroll
    for (int v = 0; v < 8; ++v) {
        int r = blockIdx.x * 16 + v + 8 * half;
        float val = acc[v] + bb;
        if (do_relu) val = fmaxf(val, 0.0f);
        out[(size_t)r * HDIM + n0 + l] = val;
    }
}

// ---------------------------------------------------------------------------
// ||w|| for the pooling score (single 64-thread block)
// ---------------------------------------------------------------------------
__global__ void wnorm_kernel(const float* __restrict__ w, float* __restrict__ wn) {
    __shared__ float s[64];
    int t = threadIdx.x;
    s[t] = w[t] * w[t];
    __syncthreads();
    for (int st = 32; st > 0; st >>= 1) {
        if (t < st) s[t] += s[t + st];
        __syncthreads();
    }
    if (t == 0) wn[0] = sqrtf(s[0]);
}

// score[i] = dot(x[i], w)/||w||  (masked-out -> -FLT_MAX)
__global__ __launch_bounds__(256) void score_kernel(const float* __restrict__ x,
                                                    const float* __restrict__ w,
                                                    const float* __restrict__ wn,
                                                    const float* __restrict__ maskf,
                                                    float* __restrict__ score) {
    __shared__ float sw[HDIM];
    if (threadIdx.x < HDIM) sw[threadIdx.x] = w[threadIdx.x];
    __syncthreads();
    int node = blockIdx.x * blockDim.x + threadIdx.x;
    if (node >= NNODE) return;
    const float* row = x + (size_t)node * HDIM;
    float acc = 0.f;
#pragma unroll 8
    for (int f = 0; f < HDIM; ++f) acc += row[f] * sw[f];
    score[node] = (maskf[node] > 0.5f) ? acc / wn[0] : NEGINF;
}

// Exact k-th largest score per graph via in-LDS bitonic sort of 2048 values.
// One block per graph, 256 threads.
__global__ __launch_bounds__(256) void kth_kernel(const float* __restrict__ score,
                                                  int k, float* __restrict__ kth) {
    __shared__ float sm[NPG];
    const int g = blockIdx.x;
    for (int i = threadIdx.x; i < NPG; i += 256) sm[i] = score[g * NPG + i];
    for (int size = 2; size <= NPG; size <<= 1) {
        for (int stride = size >> 1; stride > 0; stride >>= 1) {
            __syncthreads();
            for (int i = threadIdx.x; i < NPG; i += 256) {
                int j = i ^ stride;
                if (j > i) {
                    bool up = ((i & size) == 0);
                    float a = sm[i], b = sm[j];
                    if ((a > b) == up) { sm[i] = b; sm[j] = a; }
                }
            }
        }
    }
    __syncthreads();
    if (threadIdx.x == 0) kth[g] = sm[NPG - k];   // ascending sort -> k-th largest
}

// Gate x by tanh(score)*newmask; write new node mask.
__global__ void pool_apply(float* __restrict__ x, const float* __restrict__ score,
                           const float* __restrict__ kth, const float* __restrict__ maskOld,
                           float* __restrict__ maskNew) {
    long long t = (long long)blockIdx.x * blockDim.x + threadIdx.x;
    if (t >= (long long)NNODE * HDIM) return;
    int node = (int)(t >> 6);
    int f = (int)(t & 63);
    float sc = score[node];
    bool nm = (sc >= kth[node >> 11]) && (maskOld[node] > 0.5f);
    float gate = nm ? tanhf(sc) : 0.0f;
    x[t] *= gate;
    if (f == 0) maskNew[node] = nm ? 1.0f : 0.0f;
}

// evb &= newm[src] & newm[dst]
__global__ void edge_update(unsigned char* __restrict__ ev, const int* __restrict__ src,
                            const int* __restrict__ dst, const float* __restrict__ maskNew) {
    int e = blockIdx.x * blockDim.x + threadIdx.x;
    if (e >= NEDGE) return;
    if (ev[e] && !(maskNew[src[e]] > 0.5f && maskNew[dst[e]] > 0.5f)) ev[e] = 0;
}

// Masked mean pool per graph + final linear [64 -> 8]. One block per graph.
__global__ __launch_bounds__(256) void readout(const float* __restrict__ x,
                                               const float* __restrict__ maskf,
                                               const float* __restrict__ W,
                                               const float* __restrict__ bias,
                                               float* __restrict__ out) {
    __shared__ float sh[256];
    __shared__ float shn[4];
    __shared__ float h[HDIM];
    __shared__ float dden;
    const int g = blockIdx.x;
    const int f = threadIdx.x & 63;
    const int grp = threadIdx.x >> 6;   // 4 node groups
    float s = 0.f, n = 0.f;
    for (int node = g * NPG + grp; node < (g + 1) * NPG; node += 4) {
        float m = maskf[node];
        s += x[(size_t)node * HDIM + f] * m;
        if (f == 0) n += m;
    }
    sh[threadIdx.x] = s;
    if (f == 0) shn[grp] = n;
    __syncthreads();
    if (threadIdx.x == 0) dden = shn[0] + shn[1] + shn[2] + shn[3];
    if (grp == 0) h[f] = sh[f] + sh[f + 64] + sh[f + 128] + sh[f + 192];
    __syncthreads();
    if (threadIdx.x < CDIM) {
        int c = threadIdx.x;
        float acc = bias[c];
        for (int ff = 0; ff < HDIM; ++ff) acc += (h[ff] / dden) * W[ff * CDIM + c];
        out[g * CDIM + c] = acc;
    }
}

// ---------------------------------------------------------------------------
// Host orchestration
// ---------------------------------------------------------------------------
extern "C" void kernel_launch(void* const* d_in, const int* in_sizes, int n_in,
                              void* d_out, int out_size, void* d_ws, size_t ws_size,
                              hipStream_t stream) {
    (void)in_sizes; (void)n_in; (void)out_size; (void)ws_size;

    // ---- inputs (setup_inputs dict insertion order; params flattened in
    //      structure order: convs[i]{Wl,Wr,b}, bns[i]{g,b}, pools[0..2], lin{W,b})
    const float* x_in = (const float*)d_in[0];
    const int*   ei   = (const int*)d_in[1];
    const int*   srcI = ei;
    const int*   dstI = ei + NEDGE;
    // d_in[2] = batch (node->graph) -- implicit (node >> 11), unused
    const float *Wl[7], *Wr[7], *bcv[7], *bng[7], *bnb[7], *pools[3];
    int p = 3;
    for (int i = 0; i < 7; ++i) {
        Wl[i]  = (const float*)d_in[p++];
        Wr[i]  = (const float*)d_in[p++];
        bcv[i] = (const float*)d_in[p++];
    }
    for (int i = 0; i < 7; ++i) {
        bng[i] = (const float*)d_in[p++];
        bnb[i] = (const float*)d_in[p++];
    }
    for (int i = 0; i < 3; ++i) pools[i] = (const float*)d_in[p++];
    const float* linW = (const float*)d_in[p++];
    const float* linb = (const float*)d_in[p++];

    // ---- workspace carve (floats)
    const size_t NN64 = (size_t)NNODE * HDIM;     // 4 Mi floats
    float* wsf   = (float*)d_ws;
    float* xA    = wsf;                  // current features [NNODE, 64]
    float* xN    = xA + NN64;            // normalized features
    float* agg   = xN + NN64;            // neighbor sum accumulator
    float* cntv  = agg + NN64;           // neighbor counts  [NNODE]
    float* maskA = cntv + NNODE;
    float* maskB = maskA + NNODE;
    float* score = maskB + NNODE;
    float* stats = score + NNODE;        // 2*64+1 floats (padded to 160)
    float* kth   = stats + 160;          // [32]
    float* wn    = kth + 64;             // [1] (padded)
    unsigned char* ev = (unsigned char*)(wn + 64);   // [NEDGE] bytes

    // ---- init masks / edge-valid flags
    init_state<<<(NEDGE + 255) / 256, 256, 0, stream>>>(maskA, ev);

    float* mcur = maskA;
    float* mnext = maskB;
    const float* curX = x_in;
    int Fin = FIN0, fshift = 4;
    static const int kvals[3] = {1639, 1312, 1050};   // ceil(0.8*k) chain from 2048

    for (int ci = 0; ci < 7; ++ci) {
        // BatchNorm (masked, training-mode, biased variance)
        hipMemsetAsync(stats, 0, (2 * Fin + 1) * sizeof(float), stream);
        bn_stats<<<256, 256, 0, stream>>>(curX, mcur, Fin, stats);
        long long nEl = (long long)NNODE * Fin;
        bn_apply<<<(unsigned)((nEl + 255) / 256), 256, 0, stream>>>(
            curX, mcur, bng[ci], bnb[ci], stats, fshift, xN);

        // SAGE neighbor mean aggregation
        hipMemsetAsync(agg, 0, (size_t)NNODE * Fin * sizeof(float), stream);
        hipMemsetAsync(cntv, 0, (size_t)NNODE * sizeof(float), stream);
        long long nSc = (long long)NEDGE * Fin;
        sage_scatter<<<(unsigned)((nSc + 255) / 256), 256, 0, stream>>>(
            xN, srcI, dstI, ev, fshift, agg, cntv);

        // lin_l(mean) + lin_r(x) + b  (+ReLU except final layer) via f32 WMMA
        sage_gemm_wmma<<<NNODE / 16, 128, 0, stream>>>(
            agg, cntv, xN, Wl[ci], Wr[ci], bcv[ci], Fin, (ci != 6) ? 1 : 0, xA);

        curX = xA; Fin = HDIM; fshift = 6;

        // TopK pooling after conv pairs (ci = 1, 3, 5)
        if (ci == 1 || ci == 3 || ci == 5) {
            int blk = (ci - 1) / 2;
            wnorm_kernel<<<1, 64, 0, stream>>>(pools[blk], wn);
            score_kernel<<<NNODE / 256, 256, 0, stream>>>(xA, pools[blk], wn, mcur, score);
            kth_kernel<<<NGRAPH, 256, 0, stream>>>(score, kvals[blk], kth);
            pool_apply<<<(unsigned)(NN64 / 256), 256, 0, stream>>>(xA, score, kth, mcur, mnext);
            edge_update<<<NEDGE / 256, 256, 0, stream>>>(ev, srcI, dstI, mnext);
            float* tmp = mcur; mcur = mnext; mnext = tmp;
        }
    }

    // masked mean pool + final linear
    readout<<<NGRAPH, 256, 0, stream>>>(xA, mcur, linW, linb, (float*)d_out);
}